// AEloss_66889820668447
// MI455X (gfx1250) — compile-verified
//
#include <hip/hip_runtime.h>
#include <math.h>

// AE (associative embedding) loss for MI455X / gfx1250, wave32.
// One wave (32 lanes) per batch element: lane p in [0,30) owns person p.
// Push term's bilinear core mu_i*mu_j computed with V_WMMA_F32_16X16X4_F32
// (rank-1 outer product: only K=0 populated), 2x2 tiles covering P=30.

typedef float v2f __attribute__((ext_vector_type(2)));
typedef float v8f __attribute__((ext_vector_type(8)));

#define PMAX 30
#define KJ   17

__device__ __forceinline__ float wave_red_add(float x) {
#pragma unroll
  for (int off = 16; off > 0; off >>= 1)
    x += __shfl_xor(x, off, 32);
  return x;
}

__global__ __launch_bounds__(32) void ae_loss_kernel(
    const float* __restrict__ tags,       // [B, N] (D==1, channel 0)
    const int*   __restrict__ keypoints,  // [B, P, K, 2] int32 (idx, flag)
    float*       __restrict__ out,        // [B, 2] = (push, pull)
    int N)
{
  const int lane = threadIdx.x;          // 0..31
  const int b    = blockIdx.x;

  // ---- Phase 1: gather + per-person stats (lanes >= PMAX read person 0, masked off)
  const int p = (lane < PMAX) ? lane : 0;
  const int*   kp   = keypoints + (((size_t)b * PMAX + p) * KJ) * 2;
  const float* trow = tags + (size_t)b * N;
  const float  lane_valid = (lane < PMAX) ? 1.f : 0.f;

  float g[KJ];
  float m[KJ];
  float sum = 0.f, cnt = 0.f;
#pragma unroll
  for (int k = 0; k < KJ; ++k) {
    int idx  = kp[2 * k];
    int flag = kp[2 * k + 1];
    float mk = (flag == 1) ? lane_valid : 0.f;
    float gk = trow[idx];
    g[k] = gk;
    m[k] = mk;
    sum += gk * mk;
    cnt += mk;
  }
  const float safe_cnt = fmaxf(cnt, 1.f);
  const float mean = sum / safe_cnt;     // 0 for invalid/padded persons

  float pull_p = 0.f;
#pragma unroll
  for (int k = 0; k < KJ; ++k) {
    float d = g[k] - mean;
    pull_p += d * d * m[k];
  }
  pull_p /= safe_cnt;

  const float    pv     = (cnt > 0.f) ? 1.f : 0.f;
  const unsigned pvmask = (unsigned)__ballot(cnt > 0.f);  // wave32: low 32 bits
  const float n        = wave_red_add(pv);
  const float pull_sum = wave_red_add(pull_p * pv);

  // ---- Phase 2: push via WMMA outer product S[i][j] = mu_i * mu_j.
  // exp(-(mi-mj)^2) == exp(2*S - mi^2 - mj^2). 2x2 tiling of 16x16 over P=30.
  // A (16x4 f32): VGPR0 lanes 0-15 = column K=0 (M=lane); everything else 0.
  // B (4x16 f32): VGPR0 lanes 0-15 = row K=0 (N=lane); everything else 0.
  float push_acc = 0.f;
#pragma unroll
  for (int I = 0; I < 2; ++I) {
    float mi_bc = __shfl(mean, I * 16 + (lane & 15), 32);
    v2f A;
    A.x = (lane < 16) ? mi_bc : 0.f;
    A.y = 0.f;
#pragma unroll
    for (int J = 0; J < 2; ++J) {
      float mj_bc = __shfl(mean, J * 16 + (lane & 15), 32);
      v2f Bm;
      Bm.x = (lane < 16) ? mj_bc : 0.f;
      Bm.y = 0.f;
      v8f C = {0.f, 0.f, 0.f, 0.f, 0.f, 0.f, 0.f, 0.f};
      // 8 args: (neg_a, A, neg_b, B, c_mod, C, reuse_a, reuse_b)
      v8f D = __builtin_amdgcn_wmma_f32_16x16x4_f32(
          false, A, false, Bm, (short)0, C, false, false);

      // C/D layout: lane holds column (lane&15); VGPR v is row v + 8*(lane>=16).
      const int   ncol  = J * 16 + (lane & 15);
      const int   rbase = I * 16 + ((lane >> 4) << 3);
      const float mj    = __shfl(mean, ncol, 32);
      const float jv    = (ncol < PMAX) ? (float)((pvmask >> ncol) & 1u) : 0.f;
      const float mj2   = mj * mj;
#pragma unroll
      for (int v = 0; v < 8; ++v) {
        const int   i  = rbase + v;
        const float mi = __shfl(mean, i, 32);
        const float iv = (i < PMAX) ? (float)((pvmask >> i) & 1u) : 0.f;
        const float w  = (i < ncol) ? (iv * jv) : 0.f;   // strict upper triangle
        const float e  = 2.f * D[v] - mi * mi - mj2;     // == -(mi-mj)^2
        push_acc += w * __expf(e);
      }
    }
  }
  const float push_sum = wave_red_add(push_acc);

  if (lane == 0) {
    const float npairs = n * (n - 1.f) * 0.5f;
    float push = (n > 1.f) ? (push_sum / fmaxf(npairs, 1.f)) : push_sum;
    push *= 0.5f;
    const float pull = (n > 0.f) ? (pull_sum / fmaxf(n, 1.f)) : 0.f;
    out[2 * b + 0] = push;
    out[2 * b + 1] = pull;
  }
}

extern "C" void kernel_launch(void* const* d_in, const int* in_sizes, int n_in,
                              void* d_out, int out_size, void* d_ws, size_t ws_size,
                              hipStream_t stream) {
  const float* tags      = (const float*)d_in[0];
  const int*   keypoints = (const int*)d_in[1];
  float*       out       = (float*)d_out;

  const int B = in_sizes[1] / (PMAX * KJ * 2);  // keypoints: [B,P,K,2]
  const int N = in_sizes[0] / (B > 0 ? B : 1);  // tags: [B,N,1]

  ae_loss_kernel<<<dim3(B), dim3(32), 0, stream>>>(tags, keypoints, out, N);
}